// RTGRFID_51539607553079
// MI455X (gfx1250) — compile-verified
//
#include <hip/hip_runtime.h>
#include <stdint.h>

#define NLEN  4096
#define MLEN  4094
#define NROWS (64*128*2)
#define BLK   256
#define GRID  2048
#define RPB   (NROWS/GRID)   /* 8 rows per block */
#define LPAD  (NLEN + 8)     /* +2 floats tail pad, rounded */

__global__ __launch_bounds__(BLK) void rfid_conv1x3_async(
    const float* __restrict__ x,
    const float* __restrict__ w_rssi, const float* __restrict__ b_rssi,
    const float* __restrict__ w_phase, const float* __restrict__ b_phase,
    float* __restrict__ out)
{
    __shared__ float lds[2][LPAD];
    const int t  = threadIdx.x;
    const int r0 = blockIdx.x * RPB;

    // Both weight sets live in registers; per-row select by channel parity.
    const float rw0 = w_rssi[0],  rw1 = w_rssi[1],  rw2 = w_rssi[2],  rb = b_rssi[0];
    const float pw0 = w_phase[0], pw1 = w_phase[1], pw2 = w_phase[2], pb = b_phase[0];

    // Low 32 bits of a generic pointer to __shared__ == LDS byte offset.
    const uint32_t lbase0 = (uint32_t)(uintptr_t)&lds[0][0];
    const uint32_t lbase1 = (uint32_t)(uintptr_t)&lds[1][0];

    // Async-DMA one 16 KB row (4096 f32) into an LDS buffer: 4 x b128 per lane.
    auto stage = [&](uint32_t ldsbase, int row) {
        const float* src = x + (size_t)row * NLEN;   // uniform -> SGPR pair
        #pragma unroll
        for (int k = 0; k < 4; ++k) {
            uint32_t voff  = (uint32_t)(((k * BLK) + t) * 16);  // per-lane byte offset
            uint32_t laddr = ldsbase + voff;
            asm volatile("global_load_async_to_lds_b128 %0, %1, %2"
                         :: "v"(laddr), "v"(voff), "s"(src)
                         : "memory");
        }
    };

    stage(lbase0, r0);   // prime the pipeline

    for (int i = 0; i < RPB; ++i) {
        const int r = r0 + i;

        // My fills for buffer (i&1) are the only outstanding async ops.
        asm volatile("s_wait_asynccnt 0x0" ::: "memory");
        __syncthreads();  // all waves filled buf(i&1) AND finished reading buf((i+1)&1)

        if (i + 1 < RPB)
            stage((((i + 1) & 1) ? lbase1 : lbase0), r + 1);  // overlap fill with compute

        const int   c  = r & 1;                 // 0 = rss, 1 = phase
        const float w0 = c ? pw0 : rw0;
        const float w1 = c ? pw1 : rw1;
        const float w2 = c ? pw2 : rw2;
        const float bb = c ? pb  : rb;

        const float*  buf = &lds[i & 1][0];
        const float4* l4  = (const float4*)(buf + t * 16);  // 16B-aligned -> ds_load_b128
        float in[18];
        float4 q;
        q = l4[0]; in[0]=q.x;  in[1]=q.y;  in[2]=q.z;  in[3]=q.w;
        q = l4[1]; in[4]=q.x;  in[5]=q.y;  in[6]=q.z;  in[7]=q.w;
        q = l4[2]; in[8]=q.x;  in[9]=q.y;  in[10]=q.z; in[11]=q.w;
        q = l4[3]; in[12]=q.x; in[13]=q.y; in[14]=q.z; in[15]=q.w;
        in[16] = buf[t * 16 + 16];   // halo from LDS (t==255 reads pad; masked below)
        in[17] = buf[t * 16 + 17];

        float o[16];
        #pragma unroll
        for (int j = 0; j < 16; ++j)
            o[j] = fmaf(w2, in[j + 2], fmaf(w1, in[j + 1], fmaf(w0, in[j], bb)));

        // Output row base is 8B-aligned (r*4094*4); use b64 stores.
        float2* o2 = (float2*)(out + (size_t)r * MLEN + t * 16);
        const int nk = (t == BLK - 1) ? 7 : 8;   // last lane: 14 valid outputs
        #pragma unroll
        for (int k = 0; k < 8; ++k)
            if (k < nk) o2[k] = make_float2(o[2 * k], o[2 * k + 1]);
    }
}

extern "C" void kernel_launch(void* const* d_in, const int* in_sizes, int n_in,
                              void* d_out, int out_size, void* d_ws, size_t ws_size,
                              hipStream_t stream)
{
    const float* x       = (const float*)d_in[0];
    const float* w_rssi  = (const float*)d_in[1];
    const float* b_rssi  = (const float*)d_in[2];
    const float* w_phase = (const float*)d_in[3];
    const float* b_phase = (const float*)d_in[4];
    float* out = (float*)d_out;

    rfid_conv1x3_async<<<GRID, BLK, 0, stream>>>(x, w_rssi, b_rssi, w_phase, b_phase, out);
}